// NumDualDescriptorAB_48661979463959
// MI455X (gfx1250) — compile-verified
//
#include <hip/hip_runtime.h>

#define S_DIM 131072
#define MD    512        // M_DIM
#define LD    512        // L_DIM
#define RANK  16
#define W_DIM (S_DIM - RANK + 1)   // 131057

typedef float v2f __attribute__((ext_vector_type(2)));
typedef float v4f __attribute__((ext_vector_type(4)));
typedef float v8f __attribute__((ext_vector_type(8)));

// ---------------------------------------------------------------------------
// Kernel 0: At = Acoeff^T  (512x512, via LDS tiles -> coalesced both sides)
// ---------------------------------------------------------------------------
__global__ __launch_bounds__(256) void transpose_acoeff(
    const float* __restrict__ A, float* __restrict__ At) {
  __shared__ float tile[16][17];           // +1 pad to kill bank conflicts
  const int bx = blockIdx.x & 31;          // tile col
  const int by = blockIdx.x >> 5;          // tile row
  const int tx = threadIdx.x & 15;
  const int ty = threadIdx.x >> 4;
  tile[ty][tx] = A[(by * 16 + ty) * LD + bx * 16 + tx];
  __syncthreads();
  At[(bx * 16 + ty) * MD + by * 16 + tx] = tile[tx][ty];
}

// ---------------------------------------------------------------------------
// Kernel 1: C = Bbasis @ M  (512x512x512) with V_WMMA_F32_16X16X4_F32.
// One wave per 16x16 D tile, K-loop 512 in steps of 4 (128 WMMAs/tile).
// VGPR layouts per CDNA5 ISA 7.12.2:
//   A 16x4:  lane L, m=L&15: V0 = A[m, k0+2*half], V1 = A[m, k0+2*half+1]
//   B 4x16:  lane L, n=L&15:   V0 = B[k0+2*half, n], V1 = B[k0+2*half+1, n]
//   D 16x16: VGPR v = row (v + 8*half), col = L&15
// ---------------------------------------------------------------------------
__global__ __launch_bounds__(32) void precompute_C(
    const float* __restrict__ Bb,   // Bbasis, 512x512 row-major (j, i)
    const float* __restrict__ Mm,   // M,      512x512 row-major (i, k)
    float* __restrict__ C) {        // C,      512x512 row-major (j, k)
  const int tile = blockIdx.x;            // 0..1023
  const int tm   = (tile >> 5) * 16;      // j-tile base
  const int tn   = (tile & 31) * 16;      // k-tile base
  const int lane = threadIdx.x;           // 0..31, full wave active (EXEC all 1s)
  const int half = lane >> 4;
  const int lm   = lane & 15;

  v8f acc = {};
#pragma unroll 4
  for (int k0 = 0; k0 < MD; k0 += 4) {
    // A fragment: Bbasis[tm+lm, k0 + 2*half + {0,1}]  (contiguous float2)
    v2f a = *(const v2f*)(Bb + (tm + lm) * MD + k0 + 2 * half);
    // B fragment: M[k0 + 2*half + {0,1}, tn + lm]  (coalesced across lanes)
    v2f b;
    b.x = Mm[(k0 + 2 * half + 0) * MD + tn + lm];
    b.y = Mm[(k0 + 2 * half + 1) * MD + tn + lm];
    acc = __builtin_amdgcn_wmma_f32_16x16x4_f32(
        /*neg_a=*/false, a, /*neg_b=*/false, b,
        /*c_mod=*/(short)0, acc, /*reuse_a=*/false, /*reuse_b=*/false);
  }
#pragma unroll
  for (int v = 0; v < 8; ++v)
    C[(tm + v + 8 * half) * MD + tn + lm] = acc[v];
}

// ---------------------------------------------------------------------------
// Kernel 2: memory-bound streaming kernel.
//   V[s] = (1/16) * sum_{r<16} vec[s+r]      (register ring-free sliding sum)
//   scalar[s] = V[s] . C[s%512]              (lane-partial dot + wave32 reduce)
//   out[s,:]  = At[s%512,:] * scalar[s]
// One wave handles 64 consecutive s; lane owns 16 contiguous columns.
// Blocks are 512-aligned in s, so j = s%512 never wraps within a wave.
// ---------------------------------------------------------------------------
__global__ __launch_bounds__(256) void stream_nk(
    const float* __restrict__ vec,  // 131072 x 512
    const float* __restrict__ C,    // 512 x 512 (hot in L2)
    const float* __restrict__ At,   // 512 x 512 (hot in L2)
    float* __restrict__ out) {      // W_DIM x 512
  const int lane = threadIdx.x & 31;
  const int wave = threadIdx.x >> 5;          // 0..7
  const int s0   = blockIdx.x * 512 + wave * 64;
  const int c0   = lane * 16;                 // 16 columns = 4 x b128
  const float* vb = vec + c0;

  // init window sum over rows [s0, s0+16)
  v4f w0 = {}, w1 = {}, w2 = {}, w3 = {};
#pragma unroll
  for (int r = 0; r < RANK; ++r) {
    const v4f* p = (const v4f*)(vb + (size_t)(s0 + r) * MD);
    w0 += p[0]; w1 += p[1]; w2 += p[2]; w3 += p[3];
  }

  for (int i = 0; i < 64; ++i) {
    const int s = s0 + i;
    if (s >= W_DIM) break;                    // wave-uniform tail exit
    const int j = wave * 64 + i;              // == s % 512

    // partial dot of lane's 16 columns with C[j]
    const v4f* cj = (const v4f*)(C + (size_t)j * MD + c0);
    v4f pd = cj[0] * w0 + cj[1] * w1 + cj[2] * w2 + cj[3] * w3;
    float part = pd.x + pd.y + pd.z + pd.w;
#pragma unroll
    for (int off = 16; off > 0; off >>= 1)
      part += __shfl_xor(part, off, 32);      // wave32 butterfly reduce
    const float scalar = part * (1.0f / RANK);

    // out[s,:] = At[j,:] * scalar  (4 x b128 stores per lane)
    const v4f* aj = (const v4f*)(At + (size_t)j * MD + c0);
    v4f* o = (v4f*)(out + (size_t)s * LD + c0);
    o[0] = aj[0] * scalar;
    o[1] = aj[1] * scalar;
    o[2] = aj[2] * scalar;
    o[3] = aj[3] * scalar;

    // slide window: +row[s+16] - row[s]
    if (s + RANK < S_DIM) {
      if (s + RANK + 8 < S_DIM)
        __builtin_prefetch(vb + (size_t)(s + RANK + 8) * MD, 0, 1);
      const v4f* pin  = (const v4f*)(vb + (size_t)(s + RANK) * MD);
      const v4f* pout = (const v4f*)(vb + (size_t)s * MD);
      w0 += pin[0] - pout[0];
      w1 += pin[1] - pout[1];
      w2 += pin[2] - pout[2];
      w3 += pin[3] - pout[3];
    }
  }
}

// ---------------------------------------------------------------------------
extern "C" void kernel_launch(void* const* d_in, const int* in_sizes, int n_in,
                              void* d_out, int out_size, void* d_ws, size_t ws_size,
                              hipStream_t stream) {
  const float* vec = (const float*)d_in[0];   // vec_seq (131072 x 512)
  const float* Mm  = (const float*)d_in[1];   // M       (512 x 512)
  const float* Ac  = (const float*)d_in[2];   // Acoeff  (512 x 512)
  const float* Bb  = (const float*)d_in[3];   // Bbasis  (512 x 512)
  // d_in[4] = rank (16), hardcoded as RANK

  float* C  = (float*)d_ws;                   // 1 MB
  float* At = (float*)d_ws + (size_t)MD * LD; // 1 MB
  float* out = (float*)d_out;

  transpose_acoeff<<<1024, 256, 0, stream>>>(Ac, At);
  precompute_C<<<1024, 32, 0, stream>>>(Bb, Mm, C);
  // 256 blocks x 8 waves x 64 s-values = 131072 >= W_DIM (tail guarded)
  stream_nk<<<256, 256, 0, stream>>>(vec, C, At, out);
}